// CasualBoW_81595788690057
// MI455X (gfx1250) — compile-verified
//
#include <hip/hip_runtime.h>
#include <hip/hip_bf16.h>

// Causal bag-of-words: out[b,t,c] = mean(x[b,:t+1,c])  --  B=8, T=8192, C=512, fp32.
// Two-pass segmented scan; intra-tile cumsum done as L(16x16 lower-tri ones) @ X
// via chained V_WMMA_F32_16X16X4_F32. Memory-bandwidth bound (~256MB min traffic,
// ~11us roofline at 23.3 TB/s; x stays L2-resident between passes).

#define Bc   8
#define Tc   8192
#define Cc   512
#define NSEG 64
#define SEG  (Tc / NSEG)   // 128 rows per segment

typedef __attribute__((ext_vector_type(2))) float v2f;
typedef __attribute__((ext_vector_type(8))) float v8f;

// ---------------- Pass 1: per-(b,seg,c) segment sums -> d_ws -----------------
__global__ __launch_bounds__(256)
void CasualBoW_pass1_segsum(const float* __restrict__ x, float* __restrict__ part) {
    const int CBLK = Cc / 256;                       // 2
    int cb  = blockIdx.x % CBLK;
    int seg = (blockIdx.x / CBLK) % NSEG;
    int b   = blockIdx.x / (CBLK * NSEG);
    int c   = cb * 256 + threadIdx.x;

    const float* p = x + ((size_t)b * Tc + (size_t)seg * SEG) * Cc + c;
    float s0 = 0.f, s1 = 0.f, s2 = 0.f, s3 = 0.f;
#pragma unroll 4
    for (int i = 0; i < SEG; i += 4) {               // 16 independent loads in flight
        s0 += p[(size_t)(i + 0) * Cc];
        s1 += p[(size_t)(i + 1) * Cc];
        s2 += p[(size_t)(i + 2) * Cc];
        s3 += p[(size_t)(i + 3) * Cc];
    }
    part[((size_t)b * NSEG + seg) * Cc + c] = (s0 + s1) + (s2 + s3);
}

// ------------- Pass 2: WMMA triangular-matmul scan + scale + store -----------
__global__ __launch_bounds__(256)
void CasualBoW_pass2_scan(const float* __restrict__ x, const float* __restrict__ part,
                          float* __restrict__ out) {
    const int lane = threadIdx.x & 31;
    const int wave = threadIdx.x >> 5;               // 8 waves / block
    const int CB   = Cc / (16 * 8);                  // 4 channel-blocks per (b,seg)

    int cb  = blockIdx.x % CB;
    int seg = (blockIdx.x / CB) % NSEG;
    int b   = blockIdx.x / (CB * NSEG);

    const int cg = cb * 8 + wave;                    // 16-channel group 0..31
    const int c  = cg * 16 + (lane & 15);            // this lane's channel
    const int hi = lane >> 4;                        // lane half (WMMA layout)
    const int m  = lane & 15;                        // row index for A layout

    // Exclusive prefix over earlier segments (partials are L2-resident, 1MB).
    float off = 0.f;
    {
        const float* pp = part + (size_t)b * NSEG * Cc + c;
        for (int s = 0; s < seg; ++s) off += pp[(size_t)s * Cc];
    }

    // A-matrix chunks of the 16x16 lower-triangular ones matrix, K=4 per chunk.
    // f32 A layout (16x4): VGPR0 holds K = 0 (lanes 0-15) / 2 (lanes 16-31),
    //                      VGPR1 holds K = 1 / 3.
    v2f A[4];
#pragma unroll
    for (int kk = 0; kk < 4; ++kk) {
        int kg = kk * 4 + 2 * hi;
        A[kk].x = (m >= kg)     ? 1.f : 0.f;
        A[kk].y = (m >= kg + 1) ? 1.f : 0.f;
    }

    const size_t rowBase = (size_t)b * Tc + (size_t)seg * SEG;
    const float* xp = x   + rowBase * Cc + c;
    float*       op = out + rowBase * Cc + c;

    // Software-pipelined B-matrix tile loads (double buffer).
    v2f Bm[4];
#pragma unroll
    for (int kk = 0; kk < 4; ++kk) {
        int r0 = kk * 4 + 2 * hi;                    // B mirrors A's K striping
        Bm[kk].x = xp[(size_t)(r0 + 0) * Cc];
        Bm[kk].y = xp[(size_t)(r0 + 1) * Cc];
    }

    for (int t0 = 0; t0 < SEG; t0 += 16) {
        // Prefetch next tile's B operands before computing this tile.
        v2f Bn[4];
        if (t0 + 16 < SEG) {
#pragma unroll
            for (int kk = 0; kk < 4; ++kk) {
                int r0 = (t0 + 16) + kk * 4 + 2 * hi;
                Bn[kk].x = xp[(size_t)(r0 + 0) * Cc];
                Bn[kk].y = xp[(size_t)(r0 + 1) * Cc];
            }
        }

        // C accumulator = carry-in broadcast to all 16 rows.
        v8f acc;
#pragma unroll
        for (int r = 0; r < 8; ++r) acc[r] = off;

        // D = L @ X + carry, as 4 chained 16x16x4 f32 WMMAs.
#pragma unroll
        for (int kk = 0; kk < 4; ++kk) {
            acc = __builtin_amdgcn_wmma_f32_16x16x4_f32(
                false, A[kk], false, Bm[kk], (short)0, acc, false, false);
        }

        // Carry for next tile = inclusive row 15 = acc[7], lanes 16..31 (N=lane-16).
        off = __shfl(acc[7], 16 + (lane & 15), 32);

        // Scale by 1/(t+1): v_rcp + one Newton step (~0.5 ulp) + mul — much
        // cheaper than the IEEE divide ladder, accuracy ~1e-7 relative.
        const float cbase = (float)(seg * SEG + t0 + (hi ? 8 : 0) + 1);
#pragma unroll
        for (int r = 0; r < 8; ++r) {
            float cnt = cbase + (float)r;
            float rc  = __builtin_amdgcn_rcpf(cnt);
            rc        = rc * (2.0f - cnt * rc);      // Newton refinement
            float v   = acc[r] * rc;
            int  row  = t0 + (hi ? 8 : 0) + r;
            __builtin_nontemporal_store(v, &op[(size_t)row * Cc]);
        }

#pragma unroll
        for (int kk = 0; kk < 4; ++kk) Bm[kk] = Bn[kk];
    }
}

extern "C" void kernel_launch(void* const* d_in, const int* in_sizes, int n_in,
                              void* d_out, int out_size, void* d_ws, size_t ws_size,
                              hipStream_t stream) {
    (void)in_sizes; (void)n_in; (void)out_size; (void)ws_size;
    const float* x    = (const float*)d_in[0];
    float*       out  = (float*)d_out;
    float*       part = (float*)d_ws;                // B*NSEG*C floats = 1 MB

    // Pass 1: 8*64*2 = 1024 blocks of 256 threads.
    dim3 g1(Bc * NSEG * (Cc / 256));
    CasualBoW_pass1_segsum<<<g1, 256, 0, stream>>>(x, part);

    // Pass 2: 8*64*4 = 2048 blocks of 256 threads (8 waves x 16 channels each).
    dim3 g2(Bc * NSEG * (Cc / (16 * 8)));
    CasualBoW_pass2_scan<<<g2, 256, 0, stream>>>(x, part, out);
}